// SpectralAttentionLayer_9216999817728
// MI455X (gfx1250) — compile-verified
//
#include <hip/hip_runtime.h>
#include <hip/hip_bf16.h>
#include <math.h>

#define FD 32   // feature dim D
#define CK 96   // K*D for ChebConv concat

typedef float v2f __attribute__((ext_vector_type(2)));
typedef float v8f __attribute__((ext_vector_type(8)));

__device__ __forceinline__ float lrelu(float x, float s) { return x > 0.0f ? x : x * s; }
// order-preserving float<->uint mapping for atomicMax-based segment max
__device__ __forceinline__ unsigned f2ord(float f) {
    unsigned u = __float_as_uint(f);
    return (u & 0x80000000u) ? ~u : (u | 0x80000000u);
}
__device__ __forceinline__ float ord2f(unsigned u) {
    return (u & 0x80000000u) ? __uint_as_float(u & 0x7fffffffu) : __uint_as_float(~u);
}

// ---------------- elementwise helpers ----------------
__global__ void k_fill_f32(float* p, float v, int n) {
    int i = blockIdx.x * blockDim.x + threadIdx.x;
    if (i < n) p[i] = v;
}
__global__ void k_fill_u32(unsigned* p, unsigned v, int n) {
    int i = blockIdx.x * blockDim.x + threadIdx.x;
    if (i < n) p[i] = v;
}
__global__ void k_lrelu_inplace(float* p, int n, float slope) {
    int i = blockIdx.x * blockDim.x + threadIdx.x;
    if (i < n) p[i] = lrelu(p[i], slope);
}

// ---------------- degree / normalization ----------------
__global__ void k_degree(const int* __restrict__ dst, float* __restrict__ deg, int E) {
    int e = blockIdx.x * blockDim.x + threadIdx.x;
    if (e < E) atomicAdd(&deg[dst[e]], 1.0f);
}
__global__ void k_dinv(float* deg, int n) {   // in-place deg -> deg^-1/2 (clipped at 1)
    int i = blockIdx.x * blockDim.x + threadIdx.x;
    if (i < n) deg[i] = rsqrtf(fmaxf(deg[i], 1.0f));
}

// copy (N,32) feature matrix into slice 0 of the (N,96) concat buffer (float4 wide)
__global__ void k_copy_slice(const float* __restrict__ f, float* __restrict__ Xc, int nq) {
    int i = blockIdx.x * blockDim.x + threadIdx.x;   // nq = N*8 quads
    if (i < nq) {
        int r = i >> 3, q = (i & 7) << 2;
        *(float4*)(Xc + r * CK + q) = *(const float4*)(f + r * FD + q);
    }
}

// scatter-add: acc[dst,0:32] += X[src,0:32] * dinv[src]
// one lane per (edge, 4-feature quad): b128 gather + 4 f32 atomic adds
__global__ void k_scatter(const int* __restrict__ src, const int* __restrict__ dst,
                          const float* __restrict__ X, int xstride,
                          const float* __restrict__ dinv, float* __restrict__ acc, int E) {
    int i = blockIdx.x * blockDim.x + threadIdx.x;
    if (i >= E * 8) return;
    int e = i >> 3, q = (i & 7) << 2;
    int s = src[e], t = dst[e];
    float dv = dinv[s];
    float4 v = *(const float4*)(X + s * xstride + q);
    float* ap = acc + t * FD + q;
    atomicAdd(ap + 0, v.x * dv);
    atomicAdd(ap + 1, v.y * dv);
    atomicAdd(ap + 2, v.z * dv);
    atomicAdd(ap + 3, v.w * dv);
}

// X1 = -rn * (acc*dinv) + X0 * (rn - 1)  -> slice 1 ; also clears acc for next scatter
__global__ void k_cheb1(float* __restrict__ acc, const float* __restrict__ dinv,
                        const float* __restrict__ lam, float* __restrict__ Xc, int nd) {
    int i = blockIdx.x * blockDim.x + threadIdx.x;
    if (i >= nd) return;
    int r = i >> 5, d = i & 31;
    float rn = 2.0f / lam[0];
    float Lx = acc[i] * dinv[r];
    acc[i] = 0.0f;
    Xc[r * CK + FD + d] = -rn * Lx + Xc[r * CK + d] * (rn - 1.0f);
}
// X2 = -2*rn*(acc*dinv) + 2*(rn-1)*X1 - X0  -> slice 2 ; also clears acc
__global__ void k_cheb2(float* __restrict__ acc, const float* __restrict__ dinv,
                        const float* __restrict__ lam, float* __restrict__ Xc, int nd) {
    int i = blockIdx.x * blockDim.x + threadIdx.x;
    if (i >= nd) return;
    int r = i >> 5, d = i & 31;
    float rn = 2.0f / lam[0];
    float Lx = acc[i] * dinv[r];
    acc[i] = 0.0f;
    Xc[r * CK + 2 * FD + d] =
        -2.0f * rn * Lx + Xc[r * CK + FD + d] * (2.0f * (rn - 1.0f)) - Xc[r * CK + d];
}

// ---------------- WMMA linear: out[r,0:32] = act(X[r,0:kdim] @ W + b) ----------------
// One wave32 per 16x16 output tile; K consumed 4/step via V_WMMA_F32_16X16X4_F32.
// A 16x4: lanes 0-15 K={0,1}, lanes 16-31 K={2,3} (VGPR0/1 = the two K values).
// B 4x16: lane -> column N=lane%16, lane group selects K pair.
// C/D 16x16: VGPR g -> row M=g (lanes 0-15) / M=g+8 (lanes 16-31), N=lane%16.
__global__ void k_wmma_linear(const float* __restrict__ X, int xstride,
                              const float* __restrict__ W, const float* __restrict__ bias,
                              float* __restrict__ out, int ostride,
                              int nrows, int kdim, float slope) {
    int wave = blockIdx.x * (blockDim.x >> 5) + (threadIdx.x >> 5);
    int numRowTiles = (nrows + 15) >> 4;
    if (wave >= numRowTiles * 2) return;      // whole-wave guard: EXEC all-1s at WMMA
    int rowTile = wave >> 1;
    int ct      = wave & 1;                   // column tile (0..1 of 32 output cols)
    int lane = threadIdx.x & 31;
    int hlf  = lane >> 4;
    int mrow = lane & 15;
    int rowA = rowTile * 16 + mrow;
    int col  = ct * 16 + mrow;

    v8f acc = {};
    int ksteps = kdim >> 2;
    const float* wp = W + (hlf << 1) * FD + col;      // advances 4 rows of W per step
    if (rowTile * 16 + 16 <= nrows) {
        // full tile: uniform path, no per-load EXEC masking in the hot loop
        const float* xp = X + rowA * xstride + (hlf << 1);
        for (int k = 0; k < ksteps; ++k) {
            v2f a = *(const v2f*)xp;                  // 8B-aligned b64 load
            v2f b; b.x = wp[0]; b.y = wp[FD];
            acc = __builtin_amdgcn_wmma_f32_16x16x4_f32(
                false, a, false, b, (short)0, acc, false, false);
            xp += 4;
            wp += 4 * FD;
        }
    } else {
        for (int k = 0; k < ksteps; ++k) {
            v2f a, b;
            if (rowA < nrows) {
                const float* xp = X + rowA * xstride + (k << 2) + (hlf << 1);
                a.x = xp[0]; a.y = xp[1];
            } else {
                a.x = 0.0f; a.y = 0.0f;
            }
            b.x = wp[0]; b.y = wp[FD];
            acc = __builtin_amdgcn_wmma_f32_16x16x4_f32(
                false, a, false, b, (short)0, acc, false, false);
            wp += 4 * FD;
        }
    }
    float bv = bias[col];
#pragma unroll
    for (int g = 0; g < 8; ++g) {
        int row = rowTile * 16 + g + (hlf << 3);
        if (row < nrows) out[row * ostride + col] = lrelu(acc[g] + bv, slope);
    }
}

// ---------------- GATv2 edge kernels ----------------
// logit[e] = attn . leaky_relu(fs[src] + fd[dst], 0.2)  (one warp per edge, shuffle reduce)
__global__ void k_edge_logits(const int* __restrict__ src, const int* __restrict__ dst,
                              const float* __restrict__ fs, const float* __restrict__ fd,
                              const float* __restrict__ attn, float* __restrict__ logits, int E) {
    int i = blockIdx.x * blockDim.x + threadIdx.x;
    int e = i >> 5, d = i & 31;
    float v = 0.0f;
    if (e < E) {
        int s = src[e], t = dst[e];
        v = attn[d] * lrelu(fs[s * FD + d] + fd[t * FD + d], 0.2f);
    }
#pragma unroll
    for (int m = 16; m > 0; m >>= 1) v += __shfl_xor(v, m, 32);
    if (e < E && d == 0) logits[e] = v;
}

__global__ void k_edge_max(const int* __restrict__ dst, const float* __restrict__ logits,
                           unsigned* __restrict__ maxu, int E) {
    int e = blockIdx.x * blockDim.x + threadIdx.x;
    if (e < E) atomicMax(&maxu[dst[e]], f2ord(logits[e]));
}

__global__ void k_edge_exp(const int* __restrict__ dst, float* __restrict__ logits,
                           const unsigned* __restrict__ maxu, float* __restrict__ sumv, int E) {
    int e = blockIdx.x * blockDim.x + threadIdx.x;
    if (e < E) {
        int t = dst[e];
        float ex = __expf(logits[e] - ord2f(maxu[t]));
        logits[e] = ex;                       // reuse buffer: logits -> exp
        atomicAdd(&sumv[t], ex);
    }
}

// out[dst,0:32] += fs[src,0:32] * (exp[e] / sum[dst])   (b128 gather, quad per lane)
__global__ void k_edge_aggr(const int* __restrict__ src, const int* __restrict__ dst,
                            const float* __restrict__ fs, const float* __restrict__ ex,
                            const float* __restrict__ sumv, float* __restrict__ out, int E) {
    int i = blockIdx.x * blockDim.x + threadIdx.x;
    if (i >= E * 8) return;
    int e = i >> 3, q = (i & 7) << 2;
    int s = src[e], t = dst[e];
    float a = ex[e] / sumv[t];
    float4 v = *(const float4*)(fs + s * FD + q);
    float* op = out + t * FD + q;
    atomicAdd(op + 0, v.x * a);
    atomicAdd(op + 1, v.y * a);
    atomicAdd(op + 2, v.z * a);
    atomicAdd(op + 3, v.w * a);
}

// ---------------- launcher ----------------
extern "C" void kernel_launch(void* const* d_in, const int* in_sizes, int n_in,
                              void* d_out, int out_size, void* d_ws, size_t ws_size,
                              hipStream_t stream) {
    const int*   src  = (const int*)d_in[0];
    const int*   dst  = (const int*)d_in[1];
    const float* emb  = (const float*)d_in[2];
    const float* lam  = (const float*)d_in[3];
    const float* chW  = (const float*)d_in[4];
    const float* chb  = (const float*)d_in[5];
    const float* Ws   = (const float*)d_in[6];
    const float* bs   = (const float*)d_in[7];
    const float* Wd   = (const float*)d_in[8];
    const float* bd   = (const float*)d_in[9];
    const float* attn = (const float*)d_in[10];

    const int E = in_sizes[0];
    const int N = in_sizes[2] / FD;
    const int ND = N * FD;
    const int EQ = E * 8;    // (edge, quad) work items
    const int NQ = N * 8;
    float* out = (float*)d_out;

    // workspace partition (floats)
    float* w = (float*)d_ws;
    float* dinv   = w; w += N;
    float* Xc     = w; w += (size_t)N * CK;
    float* tmp    = w; w += (size_t)ND;      // scatter accumulator; reused as fd
    float* h1     = w; w += (size_t)ND;
    float* fs     = w; w += (size_t)ND;
    float* sumv   = w; w += N;
    float* logits = w; w += E;
    unsigned* maxu = (unsigned*)w; w += N;
    float* fd = tmp;

    const int B = 256;
    #define G(n) (((n) + B - 1) / B)
    const int wmmaWaves  = ((N + 15) / 16) * 2;
    const int wmmaBlocks = (wmmaWaves + 3) / 4;   // 128 threads = 4 waves per block

    // degree -> D^{-1/2}
    k_fill_f32<<<G(N), B, 0, stream>>>(dinv, 0.0f, N);
    k_degree<<<G(E), B, 0, stream>>>(dst, dinv, E);
    k_dinv<<<G(N), B, 0, stream>>>(dinv, N);

    // two ChebConv layers (shared weights); k_cheb1/k_cheb2 re-zero tmp after reading
    k_fill_f32<<<G(ND), B, 0, stream>>>(tmp, 0.0f, ND);
    for (int layer = 0; layer < 2; ++layer) {
        const float* featin = (layer == 0) ? emb : h1;
        k_copy_slice<<<G(NQ), B, 0, stream>>>(featin, Xc, NQ);
        k_scatter<<<G(EQ), B, 0, stream>>>(src, dst, Xc, CK, dinv, tmp, E);
        k_cheb1<<<G(ND), B, 0, stream>>>(tmp, dinv, lam, Xc, ND);
        k_scatter<<<G(EQ), B, 0, stream>>>(src, dst, Xc + FD, CK, dinv, tmp, E);
        k_cheb2<<<G(ND), B, 0, stream>>>(tmp, dinv, lam, Xc, ND);
        k_wmma_linear<<<wmmaBlocks, 128, 0, stream>>>(Xc, CK, chW, chb, h1, FD, N, CK, 0.01f);
    }

    // GATv2
    k_wmma_linear<<<wmmaBlocks, 128, 0, stream>>>(h1, FD, Ws, bs, fs, FD, N, FD, 1.0f);
    k_wmma_linear<<<wmmaBlocks, 128, 0, stream>>>(h1, FD, Wd, bd, fd, FD, N, FD, 1.0f);
    k_edge_logits<<<G(E * FD), B, 0, stream>>>(src, dst, fs, fd, attn, logits, E);
    k_fill_u32<<<G(N), B, 0, stream>>>(maxu, 0x007FFFFFu /* f2ord(-inf) */, N);
    k_edge_max<<<G(E), B, 0, stream>>>(dst, logits, maxu, E);
    k_fill_f32<<<G(N), B, 0, stream>>>(sumv, 0.0f, N);
    k_edge_exp<<<G(E), B, 0, stream>>>(dst, logits, maxu, sumv, E);
    k_fill_f32<<<G(ND), B, 0, stream>>>(out, 0.0f, ND);
    k_edge_aggr<<<G(EQ), B, 0, stream>>>(src, dst, fs, logits, sumv, out, E);
    k_lrelu_inplace<<<G(ND), B, 0, stream>>>(out, ND, 0.01f);
    #undef G
}